// TULayer_30090540876460
// MI455X (gfx1250) — compile-verified
//
#include <hip/hip_runtime.h>

typedef float v2f __attribute__((ext_vector_type(2)));
typedef float v8f __attribute__((ext_vector_type(8)));

// Problem sizes fixed by the reference.
constexpr int Bb   = 2;
constexpr int Mm   = 4096;
constexpr int Nn   = 16384;
constexpr int CIN1 = 256;
constexpr int COUT = 128;

// ---------------------------------------------------------------------------
// GEMM via V_WMMA_F32_16X16X4_F32:
//   OutT[b][p][o] = sum_c W[o][c] * X[b][c][p] + bias[o]
// One wave computes a 16(point) x 16(channel) tile; K stepped by 4.
// ---------------------------------------------------------------------------
template <int CIN, int DIM>
__global__ void convT_wmma(const float* __restrict__ X,
                           const float* __restrict__ W,
                           const float* __restrict__ bias,
                           float* __restrict__ OutT) {
  const int lane = threadIdx.x & 31;
  const int wave = threadIdx.x >> 5;
  const int tile = blockIdx.x * (blockDim.x >> 5) + wave;

  constexpr int OT = COUT / 16;      // channel tiles
  constexpr int PT = DIM / 16;       // point tiles
  const int b  = tile / (PT * OT);
  const int r0 = tile % (PT * OT);
  const int p0 = (r0 / OT) * 16;     // point-tile origin (M dim of WMMA)
  const int o0 = (r0 % OT) * 16;     // channel-tile origin (N dim of WMMA)

  const int half = lane >> 4;        // 0 -> K{0,1}, 1 -> K{2,3}
  const int l15  = lane & 15;
  const int prow = p0 + l15;         // A row (point index)
  const int ocol = o0 + l15;         // B col (output channel)

  const float* Xb = X + (size_t)b * CIN * DIM;   // [CIN][DIM]

  v8f acc;
  const float bv = bias[ocol];
#pragma unroll
  for (int r = 0; r < 8; ++r) acc[r] = bv;

#pragma unroll 4
  for (int k0 = 0; k0 < CIN; k0 += 4) {
    const int ka = k0 + half * 2;
    v2f a, w;
    a.x = Xb[(size_t)ka * DIM + prow];
    a.y = Xb[(size_t)(ka + 1) * DIM + prow];
    w.x = W[(size_t)ocol * CIN + ka];
    w.y = W[(size_t)ocol * CIN + ka + 1];
    acc = __builtin_amdgcn_wmma_f32_16x16x4_f32(false, a, false, w,
                                                (short)0, acc, false, false);
  }

  float* Ob = OutT + (size_t)b * DIM * COUT;     // [DIM][COUT]
#pragma unroll
  for (int r = 0; r < 8; ++r) {
    const int p = p0 + r + half * 8;
    Ob[(size_t)p * COUT + ocol] = acc[r];
  }
}

// ---------------------------------------------------------------------------
// Sorted top-3 insert (d0 <= d1 <= d2). Strict '<' keeps earliest index on
// ties, matching top_k tie-break.
// ---------------------------------------------------------------------------
__device__ __forceinline__ void insert3(float d, int m,
                                        float& d0, float& d1, float& d2,
                                        int& i0, int& i1, int& i2) {
  if (d < d2) {
    if (d < d1) {
      if (d < d0) {
        d2 = d1; i2 = i1; d1 = d0; i1 = i0; d0 = d; i0 = m;
      } else {
        d2 = d1; i2 = i1; d1 = d; i1 = m;
      }
    } else {
      d2 = d; i2 = m;
    }
  }
}

// ---------------------------------------------------------------------------
// KNN (k=3) with the distance cross-term on the WMMA pipe.
//   tileD(m,n) = |dst_m|^2 - 2*src_n.dst_m    (add |src_n|^2 post-selection)
// A (16 src x 4): lane&15 = src row; lanes<16 hold (dx,dy), lanes>=16 (dz,|d|^2)
// B (4 x 16 qry): lane&15 = query col; lanes<16 (-2sx,-2sy), lanes>=16 (-2sz,1)
// D: lane&15 = query col; VGPR r = src row r (+8 for lanes>=16).
// Tiles processed in groups of 4: all ds_load_b64 issued first (clause +
// immediate offsets), then 4 independent WMMAs + fmin reject filters; the
// sorted insert runs only on the rare hit path.
// Lane pair (l, l+16) covers one query's rows, merged via shfl_xor(16).
// ---------------------------------------------------------------------------
__global__ void knn_top3_wmma(const float* __restrict__ xyz1,
                              const float* __restrict__ xyz2,
                              int* __restrict__ idx3,
                              float* __restrict__ w3) {
  __shared__ float4 pts[Mm];   // 64 KB: (x, y, z, |p|^2) per source point

  const int b   = blockIdx.y;
  const int tid = threadIdx.x;

  const float* s1 = xyz1 + (size_t)b * 3 * Mm;
  for (int i = tid; i < Mm; i += blockDim.x) {
    const float x = s1[i];
    const float y = s1[Mm + i];
    const float z = s1[2 * Mm + i];
    pts[i] = make_float4(x, y, z, fmaf(x, x, fmaf(y, y, z * z)));
  }
  __syncthreads();

  const int lane = tid & 31;
  const int wave = tid >> 5;
  const int half = lane >> 4;
  const int l15  = lane & 15;
  const int n    = (blockIdx.x * (blockDim.x >> 5) + wave) * 16 + l15;

  const float* s2 = xyz2 + (size_t)b * 3 * Nn;
  const float sx = s2[n];
  const float sy = s2[Nn + n];
  const float sz = s2[2 * Nn + n];
  const float qn2 = fmaf(sx, sx, fmaf(sy, sy, sz * sz));

  v2f bvec;                                    // loop-invariant B operand
  bvec.x = half ? (-2.0f * sz) : (-2.0f * sx);
  bvec.y = half ? 1.0f         : (-2.0f * sy);

  float d0 = 3.4e38f, d1 = 3.4e38f, d2 = 3.4e38f;
  int   i0 = 0, i1 = 0, i2 = 0;

  // Per-lane base into the float4 LDS array (element units of float).
  const float* ap = (const float*)pts + (size_t)l15 * 4 + half * 2;

  for (int m0 = 0; m0 < Mm; m0 += 64) {        // 4 tiles per group
    // Issue all four A-operand loads first (ds_load_b64 x4, imm offsets).
    v2f a[4];
#pragma unroll
    for (int t = 0; t < 4; ++t)
      a[t] = *(const v2f*)(ap + (size_t)(m0 + t * 16) * 4);

#pragma unroll
    for (int t = 0; t < 4; ++t) {
      v8f c = {};
      c = __builtin_amdgcn_wmma_f32_16x16x4_f32(false, a[t], false, bvec,
                                                (short)0, c, false, false);

      // Cheap tile-level reject: min of the 8 candidates (folds to v_min3).
      const float u0 = fminf(fminf(c[0], c[1]), fminf(c[2], c[3]));
      const float u1 = fminf(fminf(c[4], c[5]), fminf(c[6], c[7]));
      const float tmin = fminf(u0, u1);

      if (tmin < d2) {               // rare: tile holds a top-3 candidate
        const int mbase = m0 + t * 16 + 8 * half;
#pragma unroll
        for (int r = 0; r < 8; ++r)
          insert3(c[r], mbase + r, d0, d1, d2, i0, i1, i2);
      }
    }
  }

  // Merge the two lane halves covering the same query.
  const float e0 = __shfl_xor(d0, 16, 32);
  const float e1 = __shfl_xor(d1, 16, 32);
  const float e2 = __shfl_xor(d2, 16, 32);
  const int   j0 = __shfl_xor(i0, 16, 32);
  const int   j1 = __shfl_xor(i1, 16, 32);
  const int   j2 = __shfl_xor(i2, 16, 32);
  insert3(e0, j0, d0, d1, d2, i0, i1, i2);
  insert3(e1, j1, d0, d1, d2, i0, i1, i2);
  insert3(e2, j2, d0, d1, d2, i0, i1, i2);

  if (half == 0) {
    const float f0 = d0 + qn2;
    const float f1 = d1 + qn2;
    const float f2 = d2 + qn2;
    float w0 = 1.0f / (f0 + 0.1f);
    float w1 = 1.0f / (f1 + 0.1f);
    float w2 = 1.0f / (f2 + 0.1f);
    const float inv = 1.0f / (w0 + w1 + w2);
    w0 *= inv; w1 *= inv; w2 *= inv;

    const size_t base = ((size_t)b * Nn + n) * 3;
    idx3[base + 0] = i0; idx3[base + 1] = i1; idx3[base + 2] = i2;
    w3[base + 0] = w0;  w3[base + 1] = w1;  w3[base + 2] = w2;
  }
}

// ---------------------------------------------------------------------------
// Gather + interpolate + residual add, writing [B][COUT][N] output layout.
// ---------------------------------------------------------------------------
__global__ void interp_add(const float* __restrict__ P1T,   // [B][M][COUT]
                           const float* __restrict__ P2T,   // [B][N][COUT]
                           const int* __restrict__ idx3,
                           const float* __restrict__ w3,
                           float* __restrict__ out2) {      // [B][COUT][N]
  const int n = blockIdx.x;
  const int b = blockIdx.y;
  const int o = threadIdx.x;

  const size_t base = ((size_t)b * Nn + n) * 3;
  const int i0 = idx3[base + 0];
  const int i1 = idx3[base + 1];
  const int i2 = idx3[base + 2];
  const float w0 = w3[base + 0];
  const float w1 = w3[base + 1];
  const float w2 = w3[base + 2];

  const float* P = P1T + (size_t)b * Mm * COUT;
  float v = w0 * P[(size_t)i0 * COUT + o];
  v = fmaf(w1, P[(size_t)i1 * COUT + o], v);
  v = fmaf(w2, P[(size_t)i2 * COUT + o], v);
  v += P2T[((size_t)b * Nn + n) * COUT + o];

  out2[((size_t)b * COUT + o) * Nn + n] = v;
}

// ---------------------------------------------------------------------------
extern "C" void kernel_launch(void* const* d_in, const int* in_sizes, int n_in,
                              void* d_out, int out_size, void* d_ws, size_t ws_size,
                              hipStream_t stream) {
  const float* xyz1 = (const float*)d_in[0];   // [B,3,M]
  const float* xyz2 = (const float*)d_in[1];   // [B,3,N]
  const float* pts1 = (const float*)d_in[2];   // [B,Cin,M]
  const float* pts2 = (const float*)d_in[3];   // [B,Cout,N]
  const float* W1   = (const float*)d_in[4];   // [Cout,Cin]
  const float* b1   = (const float*)d_in[5];   // [Cout]
  const float* W2   = (const float*)d_in[6];   // [Cout,Cout]
  const float* b2   = (const float*)d_in[7];   // [Cout]
  float* out = (float*)d_out;

  // Workspace layout
  char* ws = (char*)d_ws;
  float* P1T  = (float*)ws;                                               // B*M*COUT f32
  float* P2T  = (float*)(ws + sizeof(float) * (size_t)Bb * Mm * COUT);    // B*N*COUT f32
  char*  after = ws + sizeof(float) * ((size_t)Bb * Mm * COUT + (size_t)Bb * Nn * COUT);
  int*   idx3 = (int*)after;                                              // B*N*3 i32
  float* w3   = (float*)(after + sizeof(int) * (size_t)Bb * Nn * 3);      // B*N*3 f32

  // Output #1: xyz_2 passthrough.
  hipMemcpyAsync(out, xyz2, sizeof(float) * (size_t)Bb * 3 * Nn,
                 hipMemcpyDeviceToDevice, stream);

  // p1 = W1 @ points_1 + b1, stored transposed [B][M][COUT]
  {
    const int tiles = Bb * (Mm / 16) * (COUT / 16);   // 4096 waves
    convT_wmma<CIN1, Mm><<<tiles / 8, 256, 0, stream>>>(pts1, W1, b1, P1T);
  }
  // p2 = W2 @ points_2 + b2, stored transposed [B][N][COUT]
  {
    const int tiles = Bb * (Nn / 16) * (COUT / 16);   // 16384 waves
    convT_wmma<COUT, Nn><<<tiles / 8, 256, 0, stream>>>(pts2, W2, b2, P2T);
  }
  // 3-NN indices + inverse-distance weights (distance cross-term on WMMA)
  {
    dim3 grid(Nn / 128, Bb);                          // 8 waves * 16 queries
    knn_top3_wmma<<<grid, 256, 0, stream>>>(xyz1, xyz2, idx3, w3);
  }
  // interp + p2 -> output #2 at [B][COUT][N]
  {
    dim3 grid(Nn, Bb);
    interp_add<<<grid, COUT, 0, stream>>>(P1T, P2T, idx3, w3,
                                          out + (size_t)Bb * 3 * Nn);
  }
}